// ExtractorLoss_31327491457142
// MI455X (gfx1250) — compile-verified
//
#include <hip/hip_runtime.h>
#include <hip/hip_bf16.h>
#include <math.h>

// ---------------------------------------------------------------------------
// ExtractorLoss on MI455X (gfx1250, wave32, WMMA).
//
// PSD(f) = (sum_t x[t] cos(w f t))^2 + (sum_t x[t] sin(w f t))^2,  w = 2pi/fs.
// Split t = 256*t1 + t2 (Cooley-Tukey):
//   P[t1,f] = sum_t2 X[t1,t2] cos(w f t2)   (GEMM, v_wmma_f32_16x16x4_f32)
//   Q[t1,f] = sum_t2 X[t1,t2] sin(w f t2)   (GEMM)
//   real[f] = sum_t1 cosA*P - sinA*Q ,  imag[f] = sum_t1 sinA*P + cosA*Q,
//   A = w f 256 t1 (twiddles, exact integer phase reduction mod fs).
// ~40M naive trig evals -> ~210K trig + ~105 MFLOP of fp32 WMMA.
// ---------------------------------------------------------------------------

typedef __attribute__((ext_vector_type(2))) float v2f;
typedef __attribute__((ext_vector_type(8))) float v8f;

#define T2   256   // inner split length  (K dim of stage-A GEMM)
#define NCOL 256   // padded frequency columns (16 WMMA tiles; F=201 <= 256)

// ---------------------------------------------------------------------------
// Kernel 0: zero-padded copy of x into Xp[T1P*256] so the WMMA GEMM loop is
// completely branch-free (no EXEC manipulation around v_wmma).
// ---------------------------------------------------------------------------
__global__ void el_pad_x(const float* __restrict__ x, float* __restrict__ Xp,
                         int N, int total) {
  int i = blockIdx.x * blockDim.x + threadIdx.x;
  if (i < total) Xp[i] = (i < N) ? x[i] : 0.f;
}

// ---------------------------------------------------------------------------
// Kernel 1: basis tables, PAIR-INTERLEAVED so each lane's two K-values are
// adjacent in memory (single b64 load per fragment in the GEMM):
//   element (k, n) stored at  k*NCOL + 2*n  for even k holding (k, k+1) pairs,
//   i.e. addr(k,n) = ((k>>1)*NCOL + n)*2 + (k&1).
// Columns with f > f_max are zeroed (padding-safe).
// ---------------------------------------------------------------------------
__global__ void el_fill_basis(float* __restrict__ Bc, float* __restrict__ Bs,
                              const int* __restrict__ p_fs,
                              const int* __restrict__ p_fmin,
                              const int* __restrict__ p_fmax) {
  int idx = blockIdx.x * blockDim.x + threadIdx.x;   // 0 .. 65535
  int k = idx >> 8;                                  // t2
  int n = idx & 255;                                 // freq column
  int fsv = *p_fs;
  int f   = *p_fmin + n;
  float c = 0.f, s = 0.f;
  if (f <= *p_fmax) {
    long long ph = ((long long)f * (long long)k) % (long long)fsv; // exact mod-2pi
    float ang = (6.28318530717958647692f / (float)fsv) * (float)ph;
    sincosf(ang, &s, &c);
  }
  int pos = (((k >> 1) * NCOL + n) << 1) | (k & 1);
  Bc[pos] = c;
  Bs[pos] = s;
}

// ---------------------------------------------------------------------------
// Kernel 2: stage-A GEMMs on the fp32 matrix pipe.
// One wave (32 lanes) per 16x16 output tile; 64 k-steps of K=4; two WMMAs
// per step (cos accumulator + sin accumulator). Branch-free inner loop:
// 3x global_load_b64 + 2x v_wmma_f32_16x16x4_f32 per step.
// A 16x4 f32 layout (ISA 7.12.2): lane = M + 16*(K>=2), vgpr j <-> K = 2*(lane>>4)+j.
// B 4x16 mirrored; C/D: vgpr r -> row M = r + 8*(lane>>4), col N = lane&15.
// ---------------------------------------------------------------------------
__global__ void el_stageA_wmma(const float* __restrict__ Xp,
                               const float* __restrict__ Bc,
                               const float* __restrict__ Bs,
                               float* __restrict__ P,
                               float* __restrict__ Q) {
  const int tm   = blockIdx.x;            // row tile   (t1 / 16)
  const int tn   = blockIdx.y;            // col tile   (freq / 16)
  const int lane = threadIdx.x;           // wave32
  const int m    = lane & 15;
  const int kh   = (lane >> 4) << 1;      // 0 or 2 (K offset for this lane half)

  const int row = tm * 16 + m;            // t1 row fed by this lane (A frag)
  const int col = tn * 16 + m;            // freq col fed by this lane (B frag)

  // Per-lane base addresses; ka = k + kh is always even -> 8B-aligned b64 loads.
  const float* pa = Xp + row * T2 + kh;            // advance by 4 per k-step
  const int    bo = kh * NCOL + 2 * col;           // pair-interleaved B offset
  const float* pc = Bc + bo;                       // advance by 4*NCOL per k-step
  const float* ps = Bs + bo;

  v8f accP = {};   // cos accumulator
  v8f accQ = {};   // sin accumulator

#pragma unroll 4
  for (int k = 0; k < T2; k += 4) {
    v2f a  = *(const v2f*)(pa);
    v2f bc = *(const v2f*)(pc);
    v2f bs = *(const v2f*)(ps);
    pa += 4;
    pc += 4 * NCOL;
    ps += 4 * NCOL;
    accP = __builtin_amdgcn_wmma_f32_16x16x4_f32(false, a, false, bc,
                                                 (short)0, accP, false, false);
    accQ = __builtin_amdgcn_wmma_f32_16x16x4_f32(false, a, false, bs,
                                                 (short)0, accQ, false, false);
  }

  const int ocol  = tn * 16 + m;
  const int rbase = tm * 16 + ((lane >> 4) << 3);   // +8 rows for upper lane half
#pragma unroll
  for (int r = 0; r < 8; ++r) {
    P[(rbase + r) * NCOL + ocol] = accP[r];
    Q[(rbase + r) * NCOL + ocol] = accQ[r];
  }
}

// ---------------------------------------------------------------------------
// Kernel 3: twiddle combine + PSD + masked sums + final scalar.
// One block of 256 threads; thread j owns frequency f_min + j (F = 201 <= 256).
// Twiddle phase advanced incrementally: ph_{t1+1} = (ph_t1 + f*256) mod fs,
// exact in int arithmetic (step < fs after one reduction).
// ---------------------------------------------------------------------------
__global__ void el_finalize(const float* __restrict__ P,
                            const float* __restrict__ Q,
                            float* __restrict__ out, int T1,
                            const int* __restrict__ p_ftrue,
                            const int* __restrict__ p_fs,
                            const int* __restrict__ p_delta,
                            const int* __restrict__ p_fmin,
                            const int* __restrict__ p_fmax) {
  __shared__ float sw[256];
  __shared__ float su[256];
  const int tid  = threadIdx.x;
  const int fsv  = *p_fs;
  const int fmin = *p_fmin;
  const int fmax = *p_fmax;
  const int ftru = *p_ftrue;
  const int del  = *p_delta;
  const int F    = fmax - fmin + 1;       // assumed <= NCOL (201 here)

  float want = 0.f, unw = 0.f;
  if (tid < F) {
    const int   f = fmin + tid;
    const float w = 6.28318530717958647692f / (float)fsv;
    const int step = (int)(((long long)f * (long long)T2) % (long long)fsv);
    int ph = 0;
    float re = 0.f, im = 0.f;
    for (int t1 = 0; t1 < T1; ++t1) {
      float sA, cA;
      sincosf(w * (float)ph, &sA, &cA);
      ph += step;
      if (ph >= fsv) ph -= fsv;            // exact incremental mod-fs phase
      const float p = P[t1 * NCOL + tid];
      const float q = Q[t1 * NCOL + tid];
      re += cA * p - sA * q;               // cos(A+B) expansion
      im += sA * p + cA * q;               // sin(A+B) expansion
    }
    const float psd = re * re + im * im;
    const bool wanted = (f >= ftru - del) && (f <= ftru + del);
    want = wanted ? psd : 0.f;
    unw  = wanted ? 0.f : psd;
  }
  sw[tid] = want;
  su[tid] = unw;
  __syncthreads();
  for (int s = 128; s > 0; s >>= 1) {
    if (tid < s) { sw[tid] += sw[tid + s]; su[tid] += su[tid + s]; }
    __syncthreads();
  }
  if (tid == 0) out[0] = -10.f * log10f(sw[0] / su[0]);
}

// ---------------------------------------------------------------------------
// Host-side launcher (graph-capture safe: no malloc/sync/memcpy).
// d_in order: x(f32), f_true(i32), fs(i32), delta(i32), f_min(i32), f_max(i32).
// ---------------------------------------------------------------------------
extern "C" void kernel_launch(void* const* d_in, const int* in_sizes, int n_in,
                              void* d_out, int out_size, void* d_ws, size_t ws_size,
                              hipStream_t stream) {
  const float* x      = (const float*)d_in[0];
  const int*   f_true = (const int*)d_in[1];
  const int*   fs     = (const int*)d_in[2];
  const int*   delta  = (const int*)d_in[3];
  const int*   f_min  = (const int*)d_in[4];
  const int*   f_max  = (const int*)d_in[5];

  const int N      = in_sizes[0];
  const int T1     = (N + T2 - 1) / T2;        // 391 for N=100000
  const int tilesM = (T1 + 15) / 16;           // 25
  const int T1P    = tilesM * 16;              // 400
  const int NXP    = T1P * T2;                 // padded sample count (102400)

  // Workspace layout (floats): Xp | Bc | Bs | P | Q   (~1.75 MB total)
  float* Xp = (float*)d_ws;
  float* Bc = Xp + (size_t)NXP;
  float* Bs = Bc + (size_t)T2 * NCOL;
  float* P  = Bs + (size_t)T2 * NCOL;
  float* Q  = P  + (size_t)T1P * NCOL;

  el_pad_x<<<(NXP + 255) / 256, 256, 0, stream>>>(x, Xp, N, NXP);
  el_fill_basis<<<T2, 256, 0, stream>>>(Bc, Bs, fs, f_min, f_max);
  el_stageA_wmma<<<dim3(tilesM, NCOL / 16), 32, 0, stream>>>(Xp, Bc, Bs, P, Q);
  el_finalize<<<1, 256, 0, stream>>>(P, Q, (float*)d_out, T1,
                                     f_true, fs, delta, f_min, f_max);
}